// AdditiveContourIntegrationLayer_28424093565380
// MI455X (gfx1250) — compile-verified
//
#include <hip/hip_runtime.h>

#define C_CH 96
#define RF   25
#define HWD  55
#define MAX_TAPS     (C_CH * RF * RF)     // worst-case capacity in workspace
#define MAX_LDS_TAPS 512
#define LDS_BYTES    (384 + 384 + MAX_LDS_TAPS * 8)   // cnt[96] + off[96] + taps -> 4864 B
#define LDS_CHUNKS   (LDS_BYTES / 16)                 // 304 x b128 async transfers

typedef __attribute__((address_space(3))) unsigned char lds_uchar;

// ---------------------------------------------------------------------------
// Pass 1: build compact sparse tap table from the mask kernel.
// ws layout (bytes): [0,384) tapCount[96], [384,768) tapOffset[96],
//                    [768, 768 + MAX_TAPS*8) taps as {int packed_dydx; float w}
// ---------------------------------------------------------------------------
__global__ void build_taps_kernel(const float* __restrict__ kern,
                                  int*  __restrict__ tapCount,
                                  int*  __restrict__ tapOffset,
                                  int2* __restrict__ taps)
{
    __shared__ int cnt_s[C_CH];
    __shared__ int off_s[C_CH];
    int c = threadIdx.x;
    if (c < C_CH) {
        const float* kc = kern + c * RF * RF;
        int n = 0;
        for (int i = 0; i < RF * RF; ++i) n += (kc[i] != 0.0f) ? 1 : 0;
        cnt_s[c]    = n;
        tapCount[c] = n;
    }
    __syncthreads();
    if (c == 0) {
        int o = 0;
        for (int i = 0; i < C_CH; ++i) { off_s[i] = o; o += cnt_s[i]; }
    }
    __syncthreads();
    if (c < C_CH) {
        int o = off_s[c];
        tapOffset[c] = o;
        const float* kc = kern + c * RF * RF;
        for (int i = 0; i < RF * RF; ++i) {
            float w = kc[i];
            if (w != 0.0f) {
                int dy = i / RF - RF / 2;
                int dx = i % RF - RF / 2;
                int2 t;
                t.x = (dy << 16) | (dx & 0xFFFF);
                t.y = __float_as_int(w);
                taps[o++] = t;
            }
        }
    }
}

// ---------------------------------------------------------------------------
// Pass 2: out = x + sparse depthwise correlation.
// One float4 (4 consecutive channels of one pixel) per thread.
// Tap table is staged into LDS with gfx1250 async global->LDS copies.
// ---------------------------------------------------------------------------
__global__ void __launch_bounds__(256)
contour_sparse_kernel(const float* __restrict__ x,
                      const int*   __restrict__ ws_base,   // cnt[96], off[96], taps...
                      const int2*  __restrict__ taps_g,    // ws_base + 768 bytes
                      float*       __restrict__ out,
                      int nQuads)
{
    __shared__ __align__(16) unsigned char smem[LDS_BYTES];

    // ---- async copy of the tap table into LDS (ASYNCcnt path) ----
    unsigned lds_base = (unsigned)(unsigned long long)(lds_uchar*)&smem[0];
    for (int t = (int)threadIdx.x; t < LDS_CHUNKS; t += (int)blockDim.x) {
        unsigned loff = lds_base + (unsigned)t * 16u;
        unsigned goff = (unsigned)t * 16u;
        asm volatile("global_load_async_to_lds_b128 %0, %1, %2 offset:0"
                     :: "v"(loff), "v"(goff), "s"(ws_base)
                     : "memory");
    }
    asm volatile("s_wait_asynccnt 0" ::: "memory");
    __syncthreads();

    const int*  scnt  = (const int*)smem;
    const int*  soff  = (const int*)(smem + 384);
    const int2* staps = (const int2*)(smem + 768);

    int quad = (int)(blockIdx.x * blockDim.x + threadIdx.x);
    if (quad >= nQuads) return;

    const int CQ = C_CH / 4;
    int cq  = quad % CQ;
    int pix = quad / CQ;
    int w   = pix % HWD;
    int h   = (pix / HWD) % HWD;
    int n   = pix / (HWD * HWD);
    int c0  = cq * 4;

    float4 v = ((const float4*)x)[quad];
    float racc[4] = {v.x, v.y, v.z, v.w};

    #pragma unroll
    for (int j = 0; j < 4; ++j) {
        int c   = c0 + j;
        int cnt = scnt[c];
        if (cnt > 0) {
            int   off = soff[c];
            float s   = 0.0f;
            for (int t = 0; t < cnt; ++t) {
                int  idx = off + t;
                int2 tp  = (idx < MAX_LDS_TAPS) ? staps[idx] : taps_g[idx];
                int dy = tp.x >> 16;
                int dx = (int)(short)(tp.x & 0xFFFF);
                int hh = h + dy;
                int ww = w + dx;
                if ((unsigned)hh < (unsigned)HWD && (unsigned)ww < (unsigned)HWD) {
                    float xv = x[(size_t)((n * HWD + hh) * HWD + ww) * C_CH + c];
                    s = fmaf(xv, __int_as_float(tp.y), s);
                }
            }
            racc[j] += s;
        }
    }
    ((float4*)out)[quad] = make_float4(racc[0], racc[1], racc[2], racc[3]);
}

// ---------------------------------------------------------------------------
// Fallback (only if workspace is too small): direct dense gather, skipping
// zero weights. Correct for any mask, just slower.
// ---------------------------------------------------------------------------
__global__ void contour_dense_kernel(const float* __restrict__ x,
                                     const float* __restrict__ kern,
                                     float* __restrict__ out, int total)
{
    int idx = (int)(blockIdx.x * blockDim.x + threadIdx.x);
    if (idx >= total) return;
    int c   = idx % C_CH;
    int pix = idx / C_CH;
    int w   = pix % HWD;
    int h   = (pix / HWD) % HWD;
    int n   = pix / (HWD * HWD);
    const float* kc = kern + c * RF * RF;
    float s = x[idx];
    for (int i = 0; i < RF; ++i) {
        int hh = h + i - RF / 2;
        if ((unsigned)hh >= (unsigned)HWD) continue;
        for (int j = 0; j < RF; ++j) {
            int ww = w + j - RF / 2;
            if ((unsigned)ww >= (unsigned)HWD) continue;
            float kv = kc[i * RF + j];
            if (kv != 0.0f)
                s = fmaf(x[(size_t)((n * HWD + hh) * HWD + ww) * C_CH + c], kv, s);
        }
    }
    out[idx] = s;
}

extern "C" void kernel_launch(void* const* d_in, const int* in_sizes, int n_in,
                              void* d_out, int out_size, void* d_ws, size_t ws_size,
                              hipStream_t stream)
{
    const float* x    = (const float*)d_in[0];
    const float* kern = (const float*)d_in[1];
    float*       out  = (float*)d_out;

    int total = in_sizes[0];                 // 128*55*55*96 = 37,171,200

    size_t ws_need = 768 + (size_t)MAX_TAPS * 8 + 16;
    if (ws_size >= ws_need) {
        int*  tapCount  = (int*)d_ws;
        int*  tapOffset = tapCount + C_CH;
        int2* taps      = (int2*)((char*)d_ws + 768);

        build_taps_kernel<<<1, 128, 0, stream>>>(kern, tapCount, tapOffset, taps);

        int nQuads = total / 4;
        int blocks = (nQuads + 255) / 256;
        contour_sparse_kernel<<<blocks, 256, 0, stream>>>(
            x, (const int*)d_ws, taps, out, nQuads);
    } else {
        int blocks = (total + 255) / 256;
        contour_dense_kernel<<<blocks, 256, 0, stream>>>(x, kern, out, total);
    }
}